// GAT_65884798320874
// MI455X (gfx1250) — compile-verified
//
#include <hip/hip_runtime.h>
#include <hip/hip_bf16.h>
#include <math.h>

typedef __attribute__((ext_vector_type(2))) float v2f;
typedef __attribute__((ext_vector_type(8))) float v8f;

// ---------------------------------------------------------------------------
// Signed float atomic max via int-max (val>=0) / uint-min (val<0) trick.
// Buffer must be initialized to -inf (0xFF800000) which works for both views.
// ---------------------------------------------------------------------------
__device__ __forceinline__ void atomic_max_f32(float* addr, float val) {
  if (val >= 0.0f)
    atomicMax((int*)addr, __float_as_int(val));
  else
    atomicMin((unsigned int*)addr, __float_as_uint(val));
}

__device__ __forceinline__ float leaky02(float x) {
  return x > 0.0f ? x : 0.2f * x;
}

// ---------------------------------------------------------------------------
// FP32 WMMA GEMM: C[M,N] = A[M,K] @ B[K,N]. One wave (32 threads) per
// (16-row x NTILES*16-col) output block; blockIdx.x = row tile,
// blockIdx.y = column block of NTILES*16. Uses V_WMMA_F32_16X16X4_F32.
// M % 16 == 0, N % (NTILES*16) == 0, K % 4 == 0 (all hold here).
// ---------------------------------------------------------------------------
template <int NTILES>
__global__ void gat_gemm_wmma(const float* __restrict__ A,
                              const float* __restrict__ B,
                              float* __restrict__ C, int M, int N, int K) {
  const int mt   = blockIdx.x;
  const int nb0  = blockIdx.y * (NTILES * 16);
  const int lane = threadIdx.x & 31;
  const int half = lane >> 4;  // 0: lanes 0-15, 1: lanes 16-31
  const int l16  = lane & 15;
  const int row  = mt * 16 + l16;

  v8f acc[NTILES];
#pragma unroll
  for (int t = 0; t < NTILES; ++t) acc[t] = (v8f){0, 0, 0, 0, 0, 0, 0, 0};

  for (int k = 0; k < K; k += 4) {
    // A 16x4 fragment: lanes 0-15 hold (K=k,k+1), lanes 16-31 hold (K=k+2,k+3)
    const int kb = k + half * 2;
    const v2f a = *reinterpret_cast<const v2f*>(A + (size_t)row * K + kb);
#pragma unroll
    for (int t = 0; t < NTILES; ++t) {
      const int col = nb0 + t * 16 + l16;
      // B 4x16 fragment: V0 = rows {k, k+2}, V1 = rows {k+1, k+3}
      v2f b;
      b.x = B[(size_t)kb * N + col];
      b.y = B[(size_t)(kb + 1) * N + col];
      acc[t] = __builtin_amdgcn_wmma_f32_16x16x4_f32(
          /*neg_a=*/false, a, /*neg_b=*/false, b,
          /*c_mod=*/(short)0, acc[t], /*reuse_a=*/false, /*reuse_b=*/false);
    }
  }

  // C 16x16 layout: VGPR r -> rows (r, r+8) across lane halves
#pragma unroll
  for (int t = 0; t < NTILES; ++t) {
    const int col = nb0 + t * 16 + l16;
#pragma unroll
    for (int r = 0; r < 8; ++r) {
      const int m = mt * 16 + r + half * 8;
      C[(size_t)m * N + col] = acc[t][r];
    }
  }
}

// ---------------------------------------------------------------------------
__global__ void fill_f32(float* __restrict__ p, float v, int n) {
  int t = blockIdx.x * blockDim.x + threadIdx.x;
  if (t < n) p[t] = v;
}

// el[i,h] = sum_d z[i,h,d]*attn_l[h,d];  er (only for i < ndst) likewise.
__global__ void attn_dots(const float* __restrict__ z,
                          const float* __restrict__ attn_l,
                          const float* __restrict__ attn_r,
                          float* __restrict__ el, float* __restrict__ er,
                          int N, int ndst, int H, int D) {
  int t = blockIdx.x * blockDim.x + threadIdx.x;
  if (t >= N * H) return;
  int i = t / H, h = t % H;
  const float* zp = z + (size_t)i * H * D + (size_t)h * D;
  float sl = 0.0f, sr = 0.0f;
  for (int d = 0; d < D; ++d) {
    float v = zp[d];
    sl += v * attn_l[h * D + d];
    sr += v * attn_r[h * D + d];
  }
  el[t] = sl;
  if (i < ndst) er[i * H + h] = sr;
}

// Pass 1: per-destination segment max of leaky_relu(el[src]+er[dst])
__global__ void edge_max(const int* __restrict__ src, const int* __restrict__ dst,
                         const float* __restrict__ el, const float* __restrict__ er,
                         float* __restrict__ m, int E, int H) {
  int t = blockIdx.x * blockDim.x + threadIdx.x;
  if (t >= E * H) return;
  int e = t / H, h = t % H;
  int sv = src[e], dv = dst[e];
  float x = leaky02(el[sv * H + h] + er[dv * H + h]);
  atomic_max_f32(&m[dv * H + h], x);
}

// Pass 2: s[dst,h] += exp(e - m[dst,h])   (m == -inf => treat as 0)
__global__ void edge_sum(const int* __restrict__ src, const int* __restrict__ dst,
                         const float* __restrict__ el, const float* __restrict__ er,
                         const float* __restrict__ m, float* __restrict__ s,
                         int E, int H) {
  int t = blockIdx.x * blockDim.x + threadIdx.x;
  if (t >= E * H) return;
  int e = t / H, h = t % H;
  int sv = src[e], dv = dst[e];
  float x = leaky02(el[sv * H + h] + er[dv * H + h]);
  float mm = m[dv * H + h];
  if (mm == -__builtin_inff()) mm = 0.0f;
  atomicAdd(&s[dv * H + h], expf(x - mm));
}

// Pass 3: out[dst,h,:] += alpha * z[src,h,:]  — one wave per edge, lane == d
__global__ void edge_scatter(const int* __restrict__ src, const int* __restrict__ dst,
                             const float* __restrict__ el, const float* __restrict__ er,
                             const float* __restrict__ m, const float* __restrict__ s,
                             const float* __restrict__ z, float* __restrict__ out,
                             int E, int H, int D) {
  int wave = (blockIdx.x * blockDim.x + threadIdx.x) >> 5;
  int lane = threadIdx.x & 31;
  if (wave >= E) return;
  int sv = src[wave], dv = dst[wave];
  for (int h = 0; h < H; ++h) {
    float x = leaky02(el[sv * H + h] + er[dv * H + h]);
    float mm = m[dv * H + h];
    if (mm == -__builtin_inff()) mm = 0.0f;
    float ss = fmaxf(s[dv * H + h], 1e-9f);
    float alpha = expf(x - mm) / ss;
    if (lane < D) {
      atomicAdd(&out[(size_t)dv * H * D + (size_t)h * D + lane],
                alpha * z[(size_t)sv * H * D + (size_t)h * D + lane]);
    }
  }
}

// h[i,d] = relu( mean_h( hpre[i,h,d] + 2*b[h,d] ) )   (sum of 2 convs => 2x bias)
__global__ void mean_relu(const float* __restrict__ hpre, const float* __restrict__ b,
                          float* __restrict__ hout, int N, int H, int D) {
  int t = blockIdx.x * blockDim.x + threadIdx.x;
  if (t >= N * D) return;
  int i = t / D, d = t % D;
  float acc = 0.0f;
  for (int h = 0; h < H; ++h)
    acc += hpre[(size_t)i * H * D + (size_t)h * D + d] + 2.0f * b[h * D + d];
  acc *= (1.0f / (float)H);
  hout[t] = acc > 0.0f ? acc : 0.0f;
}

__global__ void add_bias2(float* __restrict__ out, const float* __restrict__ b,
                          int N, int D) {
  int t = blockIdx.x * blockDim.x + threadIdx.x;
  if (t >= N * D) return;
  out[t] += 2.0f * b[t % D];
}

// ---------------------------------------------------------------------------
extern "C" void kernel_launch(void* const* d_in, const int* in_sizes, int n_in,
                              void* d_out, int out_size, void* d_ws, size_t ws_size,
                              hipStream_t stream) {
  const float* feat    = (const float*)d_in[0];
  const int*   src0a   = (const int*)d_in[1];
  const int*   dst0a   = (const int*)d_in[2];
  const int*   src0b   = (const int*)d_in[3];
  const int*   dst0b   = (const int*)d_in[4];
  const int*   src1a   = (const int*)d_in[5];
  const int*   dst1a   = (const int*)d_in[6];
  const int*   src1b   = (const int*)d_in[7];
  const int*   dst1b   = (const int*)d_in[8];
  const float* W0      = (const float*)d_in[9];
  const float* attn_l0 = (const float*)d_in[10];
  const float* attn_r0 = (const float*)d_in[11];
  const float* b0      = (const float*)d_in[12];
  const float* W1      = (const float*)d_in[13];
  const float* attn_l1 = (const float*)d_in[14];
  const float* attn_r1 = (const float*)d_in[15];
  const float* b1      = (const float*)d_in[16];
  float* out = (float*)d_out;

  const int N0 = 50000, N1 = 20000, N2 = 10000;
  const int E0 = 320000, E1 = 160000;
  const int H0 = 4, D0 = 32, H1 = 1, D1 = 32;
  const int F_in = 64;

  // ---- workspace layout (floats) ----
  float* w = (float*)d_ws;
  size_t off = 0;
  float* z0   = w + off; off += (size_t)N0 * H0 * D0;   // 6.4M
  float* el0  = w + off; off += (size_t)N0 * H0;
  float* er0  = w + off; off += (size_t)N1 * H0;
  float* m0a  = w + off; off += (size_t)N1 * H0;
  float* s0a  = w + off; off += (size_t)N1 * H0;
  float* m0b  = w + off; off += (size_t)N1 * H0;
  float* s0b  = w + off; off += (size_t)N1 * H0;
  float* hpre = w + off; off += (size_t)N1 * H0 * D0;   // 2.56M
  float* h    = w + off; off += (size_t)N1 * D0;
  float* z1   = w + off; off += (size_t)N1 * H1 * D1;
  float* el1  = w + off; off += (size_t)N1 * H1;
  float* er1  = w + off; off += (size_t)N2 * H1;
  float* m1a  = w + off; off += (size_t)N2 * H1;
  float* s1a  = w + off; off += (size_t)N2 * H1;
  float* m1b  = w + off; off += (size_t)N2 * H1;
  float* s1b  = w + off; off += (size_t)N2 * H1;
  (void)ws_size; (void)in_sizes; (void)n_in; (void)out_size;

  const int TB = 256;
  auto nb = [](int n, int b) { return (n + b - 1) / b; };
  const float NEG_INF = -__builtin_inff();

  // ================= Layer 0 =================
  // z0 = feat @ W0   (M=50000, N=128, K=64) — WMMA, wave per 16x64 block
  gat_gemm_wmma<4><<<dim3(N0 / 16, 2), 32, 0, stream>>>(feat, W0, z0,
                                                        N0, H0 * D0, F_in);

  attn_dots<<<nb(N0 * H0, TB), TB, 0, stream>>>(z0, attn_l0, attn_r0, el0, er0,
                                                N0, N1, H0, D0);

  fill_f32<<<nb(N1 * H0, TB), TB, 0, stream>>>(m0a, NEG_INF, N1 * H0);
  fill_f32<<<nb(N1 * H0, TB), TB, 0, stream>>>(s0a, 0.0f, N1 * H0);
  fill_f32<<<nb(N1 * H0, TB), TB, 0, stream>>>(m0b, NEG_INF, N1 * H0);
  fill_f32<<<nb(N1 * H0, TB), TB, 0, stream>>>(s0b, 0.0f, N1 * H0);
  fill_f32<<<nb(N1 * H0 * D0, TB), TB, 0, stream>>>(hpre, 0.0f, N1 * H0 * D0);

  edge_max<<<nb(E0 * H0, TB), TB, 0, stream>>>(src0a, dst0a, el0, er0, m0a, E0, H0);
  edge_max<<<nb(E0 * H0, TB), TB, 0, stream>>>(src0b, dst0b, el0, er0, m0b, E0, H0);
  edge_sum<<<nb(E0 * H0, TB), TB, 0, stream>>>(src0a, dst0a, el0, er0, m0a, s0a, E0, H0);
  edge_sum<<<nb(E0 * H0, TB), TB, 0, stream>>>(src0b, dst0b, el0, er0, m0b, s0b, E0, H0);
  edge_scatter<<<nb(E0 * 32, TB), TB, 0, stream>>>(src0a, dst0a, el0, er0, m0a, s0a,
                                                   z0, hpre, E0, H0, D0);
  edge_scatter<<<nb(E0 * 32, TB), TB, 0, stream>>>(src0b, dst0b, el0, er0, m0b, s0b,
                                                   z0, hpre, E0, H0, D0);

  mean_relu<<<nb(N1 * D0, TB), TB, 0, stream>>>(hpre, b0, h, N1, H0, D0);

  // ================= Layer 1 =================
  // z1 = h @ W1   (M=20000, N=32, K=32) — WMMA
  gat_gemm_wmma<2><<<dim3(N1 / 16, 1), 32, 0, stream>>>(h, W1, z1,
                                                        N1, H1 * D1, D0);

  attn_dots<<<nb(N1 * H1, TB), TB, 0, stream>>>(z1, attn_l1, attn_r1, el1, er1,
                                                N1, N2, H1, D1);

  fill_f32<<<nb(N2 * H1, TB), TB, 0, stream>>>(m1a, NEG_INF, N2 * H1);
  fill_f32<<<nb(N2 * H1, TB), TB, 0, stream>>>(s1a, 0.0f, N2 * H1);
  fill_f32<<<nb(N2 * H1, TB), TB, 0, stream>>>(m1b, NEG_INF, N2 * H1);
  fill_f32<<<nb(N2 * H1, TB), TB, 0, stream>>>(s1b, 0.0f, N2 * H1);
  fill_f32<<<nb(N2 * D1, TB), TB, 0, stream>>>(out, 0.0f, N2 * D1);

  edge_max<<<nb(E1 * H1, TB), TB, 0, stream>>>(src1a, dst1a, el1, er1, m1a, E1, H1);
  edge_max<<<nb(E1 * H1, TB), TB, 0, stream>>>(src1b, dst1b, el1, er1, m1b, E1, H1);
  edge_sum<<<nb(E1 * H1, TB), TB, 0, stream>>>(src1a, dst1a, el1, er1, m1a, s1a, E1, H1);
  edge_sum<<<nb(E1 * H1, TB), TB, 0, stream>>>(src1b, dst1b, el1, er1, m1b, s1b, E1, H1);
  edge_scatter<<<nb(E1 * 32, TB), TB, 0, stream>>>(src1a, dst1a, el1, er1, m1a, s1a,
                                                   z1, out, E1, H1, D1);
  edge_scatter<<<nb(E1 * 32, TB), TB, 0, stream>>>(src1b, dst1b, el1, er1, m1b, s1b,
                                                   z1, out, E1, H1, D1);

  add_bias2<<<nb(N2 * D1, TB), TB, 0, stream>>>(out, b1, N2, D1);
}